// Pyramid_16363825398336
// MI455X (gfx1250) — compile-verified
//
#include <hip/hip_runtime.h>

// Problem constants (match reference)
#define B_   64
#define T_   4096
#define D_   512
#define C_   8
#define S_   7              // pyramid segments (1+2+4)
#define KTOT (S_ * D_)      // 3584
#define NCHUNK 16           // T chunks per batch for stage-1 streaming
#define CH (T_ / NCHUNK)    // 256 rows per chunk

typedef float v2f __attribute__((ext_vector_type(2)));
typedef float v8f __attribute__((ext_vector_type(8)));

static __device__ __forceinline__ int imin(int a, int b) { return a < b ? a : b; }
static __device__ __forceinline__ int imax(int a, int b) { return a > b ? a : b; }

// ---------------------------------------------------------------------------
// Stage 1: streaming quarter-sum partials.
// grid = (B, NCHUNK), block = 256. Thread t owns channels [2t, 2t+1] (float2).
// Each block sums its 256-row chunk into 4 quarter accumulators and writes
// partial[b][chunk][q][d]. Loads are non-temporal (x >> L2).
// ---------------------------------------------------------------------------
__global__ void seg_partial_kernel(const float* __restrict__ x,
                                   const int* __restrict__ lens,
                                   float* __restrict__ partial) {
    const int b     = blockIdx.x;
    const int chunk = blockIdx.y;
    const int tid   = threadIdx.x;          // 0..255
    const int L  = lens[b];
    const int c1 = L / 4, c2 = L / 2, c3 = (3 * L) / 4;
    const int bnd[5] = {0, c1, c2, c3, L};

    const int t0 = chunk * CH;
    const int t1 = t0 + CH;

    const float* __restrict__ xb = x + (size_t)b * T_ * D_ + 2 * tid;

#pragma unroll
    for (int q = 0; q < 4; ++q) {
        const int qs = imax(t0, bnd[q]);
        const int qe = imin(t1, bnd[q + 1]);
        v2f acc = {0.f, 0.f};
        for (int t = qs; t < qe; ++t) {
            const v2f* p = (const v2f*)(xb + (size_t)t * D_);
            acc += __builtin_nontemporal_load(p);
        }
        const size_t off = ((((size_t)b * NCHUNK + chunk) * 4 + q) * D_) + 2 * tid;
        *(v2f*)(partial + off) = acc;
    }
}

// ---------------------------------------------------------------------------
// Stage 2a: deterministic serial reduction of chunk partials -> qsum[B][4][D].
// ---------------------------------------------------------------------------
__global__ void reduce_kernel(const float* __restrict__ partial,
                              float* __restrict__ qsum) {
    const int idx = blockIdx.x * blockDim.x + threadIdx.x;  // over B*4*D = 131072
    if (idx >= B_ * 4 * D_) return;
    const int d = idx % D_;
    const int q = (idx / D_) % 4;
    const int b = idx / (4 * D_);
    float s = 0.f;
#pragma unroll
    for (int c = 0; c < NCHUNK; ++c)
        s += partial[(((size_t)b * NCHUNK + c) * 4 + q) * D_ + d];
    qsum[idx] = s;
}

// ---------------------------------------------------------------------------
// Stage 2b: 7 segment means from the 4 quarter sums. feats[B][7*D].
// Segment order matches reference: [full, half0, half1, q0, q1, q2, q3].
// ---------------------------------------------------------------------------
__global__ void feats_kernel(const float* __restrict__ qsum,
                             const int* __restrict__ lens,
                             float* __restrict__ feats) {
    const int idx = blockIdx.x * blockDim.x + threadIdx.x;  // over B*7*D = 229376
    if (idx >= B_ * S_ * D_) return;
    const int d = idx % D_;
    const int s = (idx / D_) % S_;
    const int b = idx / (S_ * D_);
    const int L = lens[b];
    const int c1 = L / 4, c2 = L / 2, c3 = (3 * L) / 4;

    const float* __restrict__ Q = qsum + (size_t)b * 4 * D_;
    const float S0 = Q[0 * D_ + d], S1 = Q[1 * D_ + d];
    const float S2 = Q[2 * D_ + d], S3 = Q[3 * D_ + d];

    float num, cnt;
    switch (s) {
        case 0: num = S0 + S1 + S2 + S3; cnt = (float)L;        break;
        case 1: num = S0 + S1;           cnt = (float)c2;       break;
        case 2: num = S2 + S3;           cnt = (float)(L - c2); break;
        case 3: num = S0;                cnt = (float)c1;       break;
        case 4: num = S1;                cnt = (float)(c2 - c1);break;
        case 5: num = S2;                cnt = (float)(c3 - c2);break;
        default:num = S3;                cnt = (float)(L - c3); break;
    }
    feats[idx] = num / cnt;
}

// ---------------------------------------------------------------------------
// Stage 3: out[64,8] = feats[64,3584] @ W^T + bias via V_WMMA_F32_16X16X4_F32.
// 1 block, 4 waves; wave w computes M-tile rows [16w, 16w+16). N padded to 16
// (cols 8..15 fed zeros, branch-free so EXEC stays all-ones for WMMA).
// A layout: lane<16 -> (M=lane, K=k0..k0+1); lane>=16 -> (M=lane-16, K=k0+2..3).
// B layout mirrors A with N in place of M. C/D: VGPR v, lane half selects M=v / v+8.
// ---------------------------------------------------------------------------
__global__ void gemm_wmma_kernel(const float* __restrict__ feats,
                                 const float* __restrict__ W,
                                 const float* __restrict__ bias,
                                 float* __restrict__ out) {
    const int wave = threadIdx.x >> 5;   // 0..3 -> M tile
    const int lane = threadIdx.x & 31;
    const int half = lane >> 4;          // 0: K=k0,k0+1 ; 1: K=k0+2,k0+3
    const int lr   = lane & 15;

    const int m = wave * 16 + lr;                    // A row (batch), always < 64
    const float nmask = (lr < C_) ? 1.0f : 0.0f;     // zero-pad B columns 8..15
    const int   ncl   = (lr < C_) ? lr : 0;

    const float* __restrict__ arow = feats + (size_t)m * KTOT;
    const float* __restrict__ bcol = W + (size_t)ncl * KTOT;  // W is [C, KTOT] row-major; B[k][n]=W[n][k]

    v8f c = {0.f, 0.f, 0.f, 0.f, 0.f, 0.f, 0.f, 0.f};
    for (int k0 = 0; k0 < KTOT; k0 += 4) {
        const int k = k0 + half * 2;
        v2f a, bb;
        a.x  = arow[k];
        a.y  = arow[k + 1];
        bb.x = bcol[k] * nmask;
        bb.y = bcol[k + 1] * nmask;
        // 8 args: (neg_a, A, neg_b, B, c_mod, C, reuse_a, reuse_b)
        c = __builtin_amdgcn_wmma_f32_16x16x4_f32(false, a, false, bb,
                                                  (short)0, c, false, false);
    }

    if (lr < C_) {
#pragma unroll
        for (int v = 0; v < 8; ++v) {
            const int row = wave * 16 + half * 8 + v;   // C/D: lane half picks M=v / v+8
            out[row * C_ + lr] = c[v] + bias[lr];
        }
    }
}

// ---------------------------------------------------------------------------
// Launch
// ---------------------------------------------------------------------------
extern "C" void kernel_launch(void* const* d_in, const int* in_sizes, int n_in,
                              void* d_out, int out_size, void* d_ws, size_t ws_size,
                              hipStream_t stream) {
    const float* x    = (const float*)d_in[0];
    const int*   lens = (const int*)d_in[1];
    const float* W    = (const float*)d_in[2];
    const float* bias = (const float*)d_in[3];
    float*       out  = (float*)d_out;

    // Workspace layout (floats): partial | qsum | feats  (~9.4 MiB total)
    float* partial = (float*)d_ws;                                   // B*NCHUNK*4*D = 2,097,152
    float* qsum    = partial + (size_t)B_ * NCHUNK * 4 * D_;         // B*4*D       =   131,072
    float* feats   = qsum    + (size_t)B_ * 4 * D_;                  // B*7*D       =   229,376

    seg_partial_kernel<<<dim3(B_, NCHUNK), 256, 0, stream>>>(x, lens, partial);

    reduce_kernel<<<(B_ * 4 * D_ + 255) / 256, 256, 0, stream>>>(partial, qsum);

    feats_kernel<<<(B_ * S_ * D_ + 255) / 256, 256, 0, stream>>>(qsum, lens, feats);

    gemm_wmma_kernel<<<1, 128, 0, stream>>>(feats, W, bias, out);
}